// Integrator_11751030521964
// MI455X (gfx1250) — compile-verified
//
#include <hip/hip_runtime.h>
#include <math.h>
#include <float.h>
#include <stdint.h>

#ifndef FLT_MAX
#define FLT_MAX 3.402823466e+38f
#endif

#define AS_GLOBAL __attribute__((address_space(1)))
#define AS_LDS    __attribute__((address_space(3)))

typedef int v4i __attribute__((vector_size(16)));

namespace {
constexpr float kNear    = 2.0f;
constexpr float kFar     = 6.0f;
constexpr float kFarDist = 1.0e10f;
constexpr int   kS       = 128;  // samples per ray
constexpr int   kWPB     = 8;    // waves per block (256 threads, wave32)
}

__device__ __forceinline__ float wave_sum32(float v) {
#pragma unroll
  for (int off = 16; off > 0; off >>= 1) v += __shfl_xor(v, off, 32);
  return v;
}

// Async copy of 16 bytes/lane from global to LDS (gfx1250 ASYNCcnt path).
__device__ __forceinline__ void async_copy_b128_to_lds(const float* gsrc, float* lds_dst) {
#if __has_builtin(__builtin_amdgcn_global_load_async_to_lds_b128)
  __builtin_amdgcn_global_load_async_to_lds_b128(
      (AS_GLOBAL v4i*)(uintptr_t)gsrc,
      (AS_LDS v4i*)(uint32_t)(uintptr_t)lds_dst,
      0, 0);
#else
  uint32_t lds_addr = (uint32_t)(uintptr_t)lds_dst;
  uint64_t gaddr    = (uint64_t)(uintptr_t)gsrc;
  asm volatile("global_load_async_to_lds_b128 %0, %1, off"
               :
               : "v"(lds_addr), "v"(gaddr)
               : "memory");
#endif
}

__device__ __forceinline__ void wait_asynccnt0() {
#if __has_builtin(__builtin_amdgcn_s_wait_asynccnt)
  __builtin_amdgcn_s_wait_asynccnt(0);
#else
  asm volatile("s_wait_asynccnt 0x0" ::: "memory");
#endif
}

__global__ __launch_bounds__(256) void nerf_integrate_kernel(
    const float* __restrict__ raw,      // (N,128,4)
    const float* __restrict__ zvals,    // (N,128)
    const float* __restrict__ rays_d,   // (N,3)
    float* __restrict__ chs_map,        // (N,3)
    float* __restrict__ depth_map,      // (N,)
    float* __restrict__ block_ent,      // (gridDim.x,) partial sums of p*logp
    int n_rays) {
  const int lane = threadIdx.x & 31;
  const int wave = threadIdx.x >> 5;
  const int ray  = blockIdx.x * kWPB + wave;
  __shared__ float s_z[kWPB * kS];  // 4 KiB: zvals staged via async-to-LDS
  __shared__ float s_ent[kWPB];

  float ent = 0.0f;  // sum of p*log(p) for this ray (negated at the end)

  if (ray < n_rays) {
    // ---- stage this ray's zvals into LDS asynchronously (512 B / wave) ----
    const float* zg = zvals + (size_t)ray * kS + (size_t)lane * 4;
    float* zl = &s_z[wave * kS + lane * 4];
    async_copy_b128_to_lds(zg, zl);

    // ---- coalesced 128-bit loads: lane l owns samples 4l..4l+3 ----
    const float4* rawv = reinterpret_cast<const float4*>(raw) +
                         (size_t)ray * kS + (size_t)lane * 4;
    float4 r0 = rawv[0];
    float4 r1 = rawv[1];
    float4 r2 = rawv[2];
    float4 r3 = rawv[3];

    // gfx1250 prefetch (global_prefetch_b8): pull next block's slices
    __builtin_prefetch((const void*)(rawv + (size_t)kWPB * kS), 0, 0);
    __builtin_prefetch((const void*)(zg + (size_t)kWPB * kS), 0, 0);

    // ---- ray direction norm (one cacheline per wave, WGP$-resident) ----
    const float* rd = rays_d + (size_t)ray * 3;
    float dx = rd[0], dy = rd[1], dz = rd[2];
    float nrm = sqrtf(dx * dx + dy * dy + dz * dz);

    // ---- consume staged zvals (this wave's own async ops only) ----
    wait_asynccnt0();
    float4 z = *reinterpret_cast<const float4*>(zl);
    // zvals[4l+4]: direct LDS read instead of a cross-lane shuffle
    int nidx = wave * kS + ((lane < 31) ? (lane * 4 + 4) : (kS - 1));
    float znext = s_z[nidx];

    float d0 = (z.y - z.x) * nrm;
    float d1 = (z.z - z.y) * nrm;
    float d2 = (z.w - z.z) * nrm;
    float d3 = (lane == 31) ? (kFarDist * nrm) : ((znext - z.w) * nrm);

    // ---- alpha = 1 - exp(-relu(sigma)*dist) ----
    float a0 = 1.0f - expf(-fmaxf(r0.w, 0.0f) * d0);
    float a1 = 1.0f - expf(-fmaxf(r1.w, 0.0f) * d1);
    float a2 = 1.0f - expf(-fmaxf(r2.w, 0.0f) * d2);
    float a3 = 1.0f - expf(-fmaxf(r3.w, 0.0f) * d3);

    float t0 = 1.0f - a0 + 1e-10f;
    float t1 = 1.0f - a1 + 1e-10f;
    float t2 = 1.0f - a2 + 1e-10f;
    float t3 = 1.0f - a3 + 1e-10f;

    // ---- wave32 exclusive scan-product of per-lane chunk products ----
    float incl = t0 * t1 * t2 * t3;
#pragma unroll
    for (int off = 1; off < 32; off <<= 1) {
      float v = __shfl_up(incl, off, 32);
      if (lane >= off) incl *= v;
    }
    float excl = __shfl_up(incl, 1, 32);
    if (lane == 0) excl = 1.0f;

    // ---- transmittance + weights for the 4 local samples ----
    float tr0 = excl;
    float tr1 = tr0 * t0;
    float tr2 = tr1 * t1;
    float tr3 = tr2 * t2;
    float w0 = a0 * tr0, w1 = a1 * tr1, w2 = a2 * tr2, w3 = a3 * tr3;

    // ---- per-lane partial accumulators ----
    float wsum = w0 + w1 + w2 + w3;
    float cr = w0 * r0.x + w1 * r1.x + w2 * r2.x + w3 * r3.x;
    float cg = w0 * r0.y + w1 * r1.y + w2 * r2.y + w3 * r3.y;
    float cb = w0 * r0.z + w1 * r1.z + w2 * r2.z + w3 * r3.z;
    const float invR = 1.0f / (kFar - kNear);
    float dsum = (w0 * (kFar - z.x) + w1 * (kFar - z.y) +
                  w2 * (kFar - z.z) + w3 * (kFar - z.w)) * invR;

    // full weight sum first (needed for the probability normalizer)
    wsum = wave_sum32(wsum);

    float plast = 1.0f - wsum + 1e-6f;
    float Zn = wsum + plast;  // sum of probs
    float invZ = 1.0f / Zn;

    {
      float p, lp;
      p = w0 * invZ; lp = fmaxf(logf(p), -FLT_MAX); ent += p * lp;
      p = w1 * invZ; lp = fmaxf(logf(p), -FLT_MAX); ent += p * lp;
      p = w2 * invZ; lp = fmaxf(logf(p), -FLT_MAX); ent += p * lp;
      p = w3 * invZ; lp = fmaxf(logf(p), -FLT_MAX); ent += p * lp;
      if (lane == 0) {
        p = plast * invZ; lp = fmaxf(logf(p), -FLT_MAX); ent += p * lp;
      }
    }

    cr   = wave_sum32(cr);
    cg   = wave_sum32(cg);
    cb   = wave_sum32(cb);
    dsum = wave_sum32(dsum);
    ent  = wave_sum32(ent);

    if (lane == 0) {
      chs_map[(size_t)ray * 3 + 0] = cr;
      chs_map[(size_t)ray * 3 + 1] = cg;
      chs_map[(size_t)ray * 3 + 2] = cb;
      depth_map[ray] = dsum / (wsum + 1e-5f);
    }
  }

  if (lane == 0) s_ent[wave] = ent;  // 0 for inactive waves
  __syncthreads();
  if (threadIdx.x == 0) {
    float e = 0.0f;
#pragma unroll
    for (int i = 0; i < kWPB; ++i) e += s_ent[i];
    block_ent[blockIdx.x] = e;
  }
}

// Deterministic fixed-tree reduction of per-block entropy partials -> scalar.
__global__ __launch_bounds__(256) void nerf_entropy_reduce_kernel(
    const float* __restrict__ block_ent, float* __restrict__ out, int n) {
  __shared__ float s[256];
  float acc = 0.0f;
  for (int i = threadIdx.x; i < n; i += 256) acc += block_ent[i];
  s[threadIdx.x] = acc;
  __syncthreads();
#pragma unroll
  for (int stride = 128; stride > 0; stride >>= 1) {
    if (threadIdx.x < stride) s[threadIdx.x] += s[threadIdx.x + stride];
    __syncthreads();
  }
  if (threadIdx.x == 0) out[0] = -s[0];  // sparsity_loss = -sum p*logp
}

extern "C" void kernel_launch(void* const* d_in, const int* in_sizes, int n_in,
                              void* d_out, int out_size, void* d_ws, size_t ws_size,
                              hipStream_t stream) {
  const float* raw    = (const float*)d_in[0];  // (N,128,4)
  const float* zvals  = (const float*)d_in[1];  // (N,128)
  const float* rays_d = (const float*)d_in[2];  // (N,3)

  const int n_rays = in_sizes[2] / 3;

  float* chs_map   = (float*)d_out;                       // N*3
  float* depth_map = chs_map + (size_t)n_rays * 3;        // N
  float* sparsity  = chs_map + (size_t)n_rays * 4;        // 1
  float* block_ent = (float*)d_ws;                        // gridDim.x floats

  const int nblocks = (n_rays + kWPB - 1) / kWPB;

  nerf_integrate_kernel<<<nblocks, 256, 0, stream>>>(
      raw, zvals, rays_d, chs_map, depth_map, block_ent, n_rays);
  nerf_entropy_reduce_kernel<<<1, 256, 0, stream>>>(block_ent, sparsity, nblocks);
}